// GCNLayer_2319282339911
// MI455X (gfx1250) — compile-verified
//
#include <hip/hip_runtime.h>

// Problem constants (match reference setup_inputs()).
#define N_NODES  50000
#define N_EDGES  1600000
#define IN_FEAT  128
#define OUT_FEAT 128

typedef __attribute__((ext_vector_type(2))) float v2f;
typedef __attribute__((ext_vector_type(8))) float v8f;

// ---------------------------------------------------------------------------
// Stage 0: zero the output (harness poisons d_out; scatter-add needs zeros).
// ---------------------------------------------------------------------------
__global__ __launch_bounds__(256) void gcn_zero_kernel(float4* __restrict__ out, int n4) {
    int i = blockIdx.x * blockDim.x + threadIdx.x;
    if (i < n4) out[i] = make_float4(0.f, 0.f, 0.f, 0.f);
}

// ---------------------------------------------------------------------------
// Stage 1: H = X @ W^T  (fp32 GEMM via V_WMMA_F32_16X16X4_F32).
//
// One wave computes one 16(M=rows of X) x 16(N=out features) tile of H,
// accumulating over K=128 in steps of 4 (32 WMMAs, fully unrolled).
//
// A-fragment (16x4 f32, ISA layout): lanes 0-15 hold M=lane, K={k0,k0+1};
//   lanes 16-31 hold M=lane-16, K={k0+2,k0+3}.  -> contiguous v2f load.
// B-fragment (4x16 f32): B[k][n] = W[o0+n][k] (since H = X * W^T); same
//   lane-half K split -> contiguous v2f load from W row (o0+ln).
// C/D (16x16 f32, 8 VGPRs): VGPR v -> row rbase+v, col o0+ln, where
//   rbase = row0 + (lane<16 ? 0 : 8).
// ---------------------------------------------------------------------------
__global__ __launch_bounds__(256) void gcn_gemm_kernel(const float* __restrict__ X,
                                                       const float* __restrict__ W,
                                                       float* __restrict__ H) {
    const int lane = threadIdx.x & 31;
    const int wave = threadIdx.x >> 5;

    const int row0 = blockIdx.x * 16;   // 50000 / 16 = 3125 blocks, no tail
    const int o0   = wave * 16;         // 8 waves cover 128 out features
    const int ln   = lane & 15;
    const int koff = (lane >> 4) << 1;  // 0 for lanes 0-15, 2 for lanes 16-31

    const float* __restrict__ xa = X + (size_t)(row0 + ln) * IN_FEAT + koff;
    const float* __restrict__ wb = W + (size_t)(o0 + ln) * IN_FEAT + koff;

    v8f acc = {};
#pragma unroll
    for (int k0 = 0; k0 < IN_FEAT; k0 += 4) {
        v2f a = *(const v2f*)(xa + k0);   // global_load_b64, imm offset
        v2f b = *(const v2f*)(wb + k0);   // global_load_b64, imm offset
        // 8 args: (neg_a, A, neg_b, B, c_mod, C, reuse_a, reuse_b)
        acc = __builtin_amdgcn_wmma_f32_16x16x4_f32(
            false, a, false, b, (short)0, acc, false, false);
    }

    const int rbase = row0 + ((lane >> 4) << 3);
    float* __restrict__ hp = H + (size_t)rbase * OUT_FEAT + o0 + ln;
#pragma unroll
    for (int v = 0; v < 8; ++v)
        hp[(size_t)v * OUT_FEAT] = acc[v];
}

// ---------------------------------------------------------------------------
// Stage 2: out = A @ H via COO scatter-add.  One wave32 per edge: lane l
// handles features [4l, 4l+3].  H row (512B) gathered as coalesced float4;
// 4x hardware global_atomic_add_f32 into out row.  H (25.6 MB) and out
// (25.6 MB) both fit in the 192 MB L2, so this runs at L2 atomic rate.
// ---------------------------------------------------------------------------
__global__ __launch_bounds__(256) void gcn_spmm_kernel(const float* __restrict__ H,
                                                       const float* __restrict__ vals,
                                                       const int* __restrict__ rows,
                                                       const int* __restrict__ cols,
                                                       float* __restrict__ out) {
    const int e    = (int)((blockIdx.x * blockDim.x + threadIdx.x) >> 5);
    const int lane = threadIdx.x & 31;
    if (e >= N_EDGES) return;

    const int   r = rows[e];
    const int   c = cols[e];
    const float v = vals[e];

    const float4 h = ((const float4*)(H + (size_t)c * OUT_FEAT))[lane];
    float* __restrict__ o = out + (size_t)r * OUT_FEAT + lane * 4;

    unsafeAtomicAdd(o + 0, v * h.x);   // global_atomic_add_f32 (no CAS loop)
    unsafeAtomicAdd(o + 1, v * h.y);
    unsafeAtomicAdd(o + 2, v * h.z);
    unsafeAtomicAdd(o + 3, v * h.w);
}

// ---------------------------------------------------------------------------
// Launch: zero(out) -> GEMM (H into d_ws, 25.6 MB) -> SpMM scatter-add.
// Stream ordering serializes the dependency H -> SpMM.
// ---------------------------------------------------------------------------
extern "C" void kernel_launch(void* const* d_in, const int* in_sizes, int n_in,
                              void* d_out, int out_size, void* d_ws, size_t ws_size,
                              hipStream_t stream) {
    const float* X      = (const float*)d_in[0];
    const float* W      = (const float*)d_in[1];
    const float* A_vals = (const float*)d_in[2];
    const int*   A_rows = (const int*)d_in[3];
    const int*   A_cols = (const int*)d_in[4];
    float*       out    = (float*)d_out;
    float*       H      = (float*)d_ws;   // 50000*128*4 = 25.6 MB scratch

    const int n4 = N_NODES * OUT_FEAT / 4;
    gcn_zero_kernel<<<(n4 + 255) / 256, 256, 0, stream>>>((float4*)out, n4);

    gcn_gemm_kernel<<<N_NODES / 16, 256, 0, stream>>>(X, W, H);

    gcn_spmm_kernel<<<N_EDGES / 8, 256, 0, stream>>>(H, A_vals, A_rows, A_cols, out);
}